// GATLayer_48000554500594
// MI455X (gfx1250) — compile-verified
//
#include <hip/hip_runtime.h>

// ---------------------------------------------------------------------------
// TransformerConv layer for MI455X (gfx1250, wave32, WMMA).
// fp32 WMMA (V_WMMA_F32_16X16X4_F32) with WEIGHT-STATIONARY persistent waves:
// each wave holds the B fragments for 2 N-tiles (128 VGPRs) in registers and
// grid-strides over row tiles, so the steady-state inner loop is 64 v_wmma
// fed only by 32 A-loads -- no per-WMMA weight reloads from L2.
// ---------------------------------------------------------------------------

typedef float v2f __attribute__((ext_vector_type(2)));
typedef float v8f __attribute__((ext_vector_type(8)));

#define DD 128
#define HH 8

// D = A(16x4) * B(4x16) + C, fp32.
__device__ __forceinline__ v8f wmma4(v2f a, v2f b, v8f c) {
  return __builtin_amdgcn_wmma_f32_16x16x4_f32(false, a, false, b, (short)0, c,
                                               false, false);
}

// A fragments for a 16x128 row tile (row-major, ld = 128).
// lanes 0-15: M=lane, K={4t,4t+1}; lanes 16-31: M=lane-16, K={4t+2,4t+3}.
__device__ __forceinline__ void load_A(const float* __restrict__ base, int row0,
                                       v2f a[32]) {
  const int lane = threadIdx.x & 31;
  const int l = lane & 15;
  const int half = lane >> 4;
  const float* p = base + (size_t)(row0 + l) * DD + 2 * half;
#pragma unroll
  for (int t = 0; t < 32; ++t)
    a[t] = *(const v2f*)(p + 4 * t);
}

// Weight-stationary B fragments for 2 consecutive N-tiles (n0, n0+1) of
// row-major W[128][128].  VGPR0 = K row (2*half), VGPR1 = K row (2*half+1),
// N striped across 16 lanes.  128 VGPRs, loaded once per kernel.
__device__ __forceinline__ void load_B2(const float* __restrict__ W, int n0,
                                        v2f b[2][32]) {
  const int lane = threadIdx.x & 31;
  const int l = lane & 15;
  const int half = lane >> 4;
#pragma unroll
  for (int nn = 0; nn < 2; ++nn) {
    const float* wp = W + (size_t)(2 * half) * DD + (n0 + nn) * 16 + l;
#pragma unroll
    for (int t = 0; t < 32; ++t) {
      b[nn][t].x = wp[(4 * t) * DD];
      b[nn][t].y = wp[(4 * t + 1) * DD];
    }
  }
}

// Store C/D tile: VGPR r holds row (r + 8*half), col = col0 + (lane&15).
__device__ __forceinline__ void store_D(float* __restrict__ out, int row0,
                                        int col0, v8f acc) {
  const int lane = threadIdx.x & 31;
  const int l = lane & 15;
  const int half = lane >> 4;
#pragma unroll
  for (int r = 0; r < 8; ++r)
    out[(size_t)(row0 + r + 8 * half) * DD + col0 + l] = acc[r];
}

// ---------------------------------------------------------------------------
// K1: {q,k,v,skip} = x @ {Wq,Wk,Wv,Wskip} + bias.
// gridDim.y in [0,4) selects the matrix; 4 waves x 2 N-tiles = all 8 tiles.
// Persistent: grid-stride over row tiles with B in registers.
// ---------------------------------------------------------------------------
__global__ __launch_bounds__(128) void k_qkvs(
    const float* __restrict__ x, const float* __restrict__ Wq,
    const float* __restrict__ bq, const float* __restrict__ Wk,
    const float* __restrict__ bk, const float* __restrict__ Wv,
    const float* __restrict__ bv, const float* __restrict__ Ws,
    const float* __restrict__ bs, float* __restrict__ q, float* __restrict__ k,
    float* __restrict__ v, float* __restrict__ s, int rowTiles) {
  const int wave = threadIdx.x >> 5;
  const int n0 = wave * 2;

  const float* W;
  const float* bias;
  float* o;
  if (blockIdx.y == 0)      { W = Wq; bias = bq; o = q; }
  else if (blockIdx.y == 1) { W = Wk; bias = bk; o = k; }
  else if (blockIdx.y == 2) { W = Wv; bias = bv; o = v; }
  else                      { W = Ws; bias = bs; o = s; }

  v2f b[2][32];
  load_B2(W, n0, b);

  const int l = threadIdx.x & 15;
  const float bias0 = bias[n0 * 16 + l];
  const float bias1 = bias[(n0 + 1) * 16 + l];

  for (int tile = blockIdx.x; tile < rowTiles; tile += gridDim.x) {
    const int row0 = tile * 16;
    v2f a[32];
    load_A(x, row0, a);

    v8f acc0, acc1;
#pragma unroll
    for (int r = 0; r < 8; ++r) { acc0[r] = bias0; acc1[r] = bias1; }
#pragma unroll
    for (int t = 0; t < 32; ++t) {
      acc0 = wmma4(a[t], b[0][t], acc0);
      acc1 = wmma4(a[t], b[1][t], acc1);
    }
    store_D(o, row0, n0 * 16, acc0);
    store_D(o, row0, (n0 + 1) * 16, acc1);
  }
}

// ---------------------------------------------------------------------------
// K2: e = edge_attr @ We (no bias) + fused attention logits:
// alpha[edge,h] = dot(q[dst,h,:], k[src,h,:] + e[edge,h,:]) / sqrt(16).
// Wave owns heads {2w, 2w+1} (head h == N-tile h); C=16 dot is a 16-lane
// shuffle reduction on the WMMA accumulator stripes.
// ---------------------------------------------------------------------------
__global__ __launch_bounds__(128) void k_edge(
    const long long* __restrict__ ei, const float* __restrict__ eattr,
    const float* __restrict__ We, const float* __restrict__ q,
    const float* __restrict__ k, float* __restrict__ e,
    float* __restrict__ alpha, int E, int edgeTiles) {
  const int wave = threadIdx.x >> 5;
  const int h0 = wave * 2;

  v2f b[2][32];
  load_B2(We, h0, b);

  const long long* srcI = ei;      // edge_index[0] = src (j)
  const long long* dstI = ei + E;  // edge_index[1] = dst (i)

  const int lane = threadIdx.x & 31;
  const int l = lane & 15;
  const int half = lane >> 4;

  for (int tile = blockIdx.x; tile < edgeTiles; tile += gridDim.x) {
    const int e0 = tile * 16;
    v2f a[32];
    load_A(eattr, e0, a);

    int se[8], de[8];
#pragma unroll
    for (int r = 0; r < 8; ++r) {
      const int edge = e0 + r + 8 * half;
      se[r] = (int)srcI[edge];
      de[r] = (int)dstI[edge];
    }

    v8f acc[2];
#pragma unroll
    for (int r = 0; r < 8; ++r) { acc[0][r] = 0.0f; acc[1][r] = 0.0f; }
#pragma unroll
    for (int t = 0; t < 32; ++t) {
      acc[0] = wmma4(a[t], b[0][t], acc[0]);
      acc[1] = wmma4(a[t], b[1][t], acc[1]);
    }
    store_D(e, e0, h0 * 16, acc[0]);
    store_D(e, e0, (h0 + 1) * 16, acc[1]);

#pragma unroll
    for (int hh = 0; hh < 2; ++hh) {
      const int h = h0 + hh;
#pragma unroll
      for (int r = 0; r < 8; ++r) {
        const float kv = k[(size_t)se[r] * DD + h * 16 + l];
        const float qv = q[(size_t)de[r] * DD + h * 16 + l];
        float p = qv * (kv + acc[hh][r]);
        p += __shfl_xor(p, 1, 32);
        p += __shfl_xor(p, 2, 32);
        p += __shfl_xor(p, 4, 32);
        p += __shfl_xor(p, 8, 32);
        if (l == 0)
          alpha[(size_t)(e0 + r + 8 * half) * HH + h] = p * 0.25f;
      }
    }
  }
}

// ---------------------------------------------------------------------------
// Scatter softmax: monotone float<->int key for atomicMax.
// ---------------------------------------------------------------------------
__device__ __forceinline__ int fkey(float f) {
  int i = __float_as_int(f);
  return i >= 0 ? i : (i ^ 0x7fffffff);
}
__device__ __forceinline__ float funkey(int i) {
  return __int_as_float(i >= 0 ? i : (i ^ 0x7fffffff));
}

__global__ __launch_bounds__(256) void k_init(int* __restrict__ amax,
                                              float* __restrict__ denom,
                                              float* __restrict__ agg, int N) {
  const size_t tid = (size_t)blockIdx.x * blockDim.x + threadIdx.x;
  if (tid < (size_t)N * HH) {
    amax[tid] = (int)0x80000000;
    denom[tid] = 0.0f;
  }
  if (tid < (size_t)N * DD) agg[tid] = 0.0f;
}

__global__ __launch_bounds__(256) void k_amax(const float* __restrict__ alpha,
                                              const long long* __restrict__ ei,
                                              int* __restrict__ amax, int E) {
  const size_t tid = (size_t)blockIdx.x * blockDim.x + threadIdx.x;
  if (tid >= (size_t)E * HH) return;
  const int edge = (int)(tid >> 3);
  const int h = (int)(tid & 7);
  const int d = (int)ei[E + edge];
  atomicMax(&amax[d * HH + h], fkey(alpha[tid]));
}

__global__ __launch_bounds__(256) void k_softmax(
    float* __restrict__ alpha, const long long* __restrict__ ei,
    const int* __restrict__ amax, float* __restrict__ denom, int E) {
  const size_t tid = (size_t)blockIdx.x * blockDim.x + threadIdx.x;
  if (tid >= (size_t)E * HH) return;
  const int edge = (int)(tid >> 3);
  const int h = (int)(tid & 7);
  const int d = (int)ei[E + edge];
  const float m = funkey(amax[d * HH + h]);
  const float ex = expf(alpha[tid] - m);
  alpha[tid] = ex;
  atomicAdd(&denom[d * HH + h], ex);
}

// agg[dst, c] += (ex/denom) * (v[src, c] + e[edge, c])
__global__ __launch_bounds__(256) void k_msg(
    const float* __restrict__ alpha, const float* __restrict__ denom,
    const long long* __restrict__ ei, const float* __restrict__ v,
    const float* __restrict__ e, float* __restrict__ agg, int E) {
  const size_t tid = (size_t)blockIdx.x * blockDim.x + threadIdx.x;
  if (tid >= (size_t)E * DD) return;
  const int edge = (int)(tid >> 7);
  const int c = (int)(tid & 127);
  const int h = c >> 4;
  const int s = (int)ei[edge];
  const int d = (int)ei[E + edge];
  const float w = alpha[(size_t)edge * HH + h] / (denom[d * HH + h] + 1e-16f);
  const float val = w * (v[(size_t)s * DD + c] + e[tid]);
  atomicAdd(&agg[(size_t)d * DD + c], val);
}

// ---------------------------------------------------------------------------
// LayerNorm + residual, one wave per node (4 cols/lane, shuffle reductions).
// out[node] = res[node] + LN(in1[node] (+ in2[node]))
// ---------------------------------------------------------------------------
template <bool TWO_IN>
__global__ __launch_bounds__(256) void k_ln(const float* __restrict__ res,
                                            const float* __restrict__ in1,
                                            const float* __restrict__ in2,
                                            const float* __restrict__ g,
                                            const float* __restrict__ be,
                                            float* __restrict__ out, int N) {
  const int node = blockIdx.x * 8 + (threadIdx.x >> 5);
  if (node >= N) return;
  const int lane = threadIdx.x & 31;

  float t[4];
  float sum = 0.0f;
#pragma unroll
  for (int j = 0; j < 4; ++j) {
    const int c = lane + 32 * j;
    float x = in1[(size_t)node * DD + c];
    if (TWO_IN) x += in2[(size_t)node * DD + c];
    t[j] = x;
    sum += x;
  }
#pragma unroll
  for (int m = 16; m >= 1; m >>= 1) sum += __shfl_xor(sum, m, 32);
  const float mu = sum * (1.0f / 128.0f);

  float vs = 0.0f;
#pragma unroll
  for (int j = 0; j < 4; ++j) {
    const float dlt = t[j] - mu;
    vs += dlt * dlt;
  }
#pragma unroll
  for (int m = 16; m >= 1; m >>= 1) vs += __shfl_xor(vs, m, 32);
  const float rstd = rsqrtf(vs * (1.0f / 128.0f) + 1e-5f);

#pragma unroll
  for (int j = 0; j < 4; ++j) {
    const int c = lane + 32 * j;
    out[(size_t)node * DD + c] =
        res[(size_t)node * DD + c] + (t[j] - mu) * rstd * g[c] + be[c];
  }
}

// ---------------------------------------------------------------------------
// FFN GEMM: out = (silu?)(in @ W + b).  Weight-stationary, wave owns 2 N-tiles.
// ---------------------------------------------------------------------------
template <bool SILU>
__global__ __launch_bounds__(128) void k_ffn(const float* __restrict__ in,
                                             const float* __restrict__ W,
                                             const float* __restrict__ bias,
                                             float* __restrict__ out,
                                             int rowTiles) {
  const int wave = threadIdx.x >> 5;
  const int n0 = wave * 2;

  v2f b[2][32];
  load_B2(W, n0, b);

  const int l = threadIdx.x & 15;
  const float bias0 = bias[n0 * 16 + l];
  const float bias1 = bias[(n0 + 1) * 16 + l];

  for (int tile = blockIdx.x; tile < rowTiles; tile += gridDim.x) {
    const int row0 = tile * 16;
    v2f a[32];
    load_A(in, row0, a);

    v8f acc0, acc1;
#pragma unroll
    for (int r = 0; r < 8; ++r) { acc0[r] = bias0; acc1[r] = bias1; }
#pragma unroll
    for (int t = 0; t < 32; ++t) {
      acc0 = wmma4(a[t], b[0][t], acc0);
      acc1 = wmma4(a[t], b[1][t], acc1);
    }
    if (SILU) {
#pragma unroll
      for (int r = 0; r < 8; ++r) {
        float z = acc0[r];
        acc0[r] = z / (1.0f + expf(-z));
        z = acc1[r];
        acc1[r] = z / (1.0f + expf(-z));
      }
    }
    store_D(out, row0, n0 * 16, acc0);
    store_D(out, row0, (n0 + 1) * 16, acc1);
  }
}

// ---------------------------------------------------------------------------
extern "C" void kernel_launch(void* const* d_in, const int* in_sizes, int n_in,
                              void* d_out, int out_size, void* d_ws,
                              size_t ws_size, hipStream_t stream) {
  const long long* ei = (const long long*)d_in[0];  // [2,E] int64
  const float* x      = (const float*)d_in[1];      // [N,128]
  const float* eattr  = (const float*)d_in[2];      // [E,128]
  const float* Wq = (const float*)d_in[3];
  const float* bq = (const float*)d_in[4];
  const float* Wk = (const float*)d_in[5];
  const float* bk = (const float*)d_in[6];
  const float* Wv = (const float*)d_in[7];
  const float* bv = (const float*)d_in[8];
  const float* We = (const float*)d_in[9];
  const float* Ws = (const float*)d_in[10];
  const float* bs = (const float*)d_in[11];
  const float* W1 = (const float*)d_in[12];
  const float* b1 = (const float*)d_in[13];
  const float* W2 = (const float*)d_in[14];
  const float* b2 = (const float*)d_in[15];
  const float* g1  = (const float*)d_in[16];
  const float* be1 = (const float*)d_in[17];
  const float* g2  = (const float*)d_in[18];
  const float* be2 = (const float*)d_in[19];

  const int E = in_sizes[0] / 2;
  const int N = in_sizes[1] / DD;
  const int rowTiles = N / 16;   // 6250 (N % 16 == 0)
  const int edgeTiles = E / 16;  // 37500 (E % 16 == 0)

  const size_t ND = (size_t)N * DD;
  const size_t ED = (size_t)E * DD;
  const size_t NH = (size_t)N * HH;
  const size_t EH = (size_t)E * HH;

  float* ws = (float*)d_ws;
  float* q     = ws;
  float* k     = q + ND;
  float* v     = k + ND;
  float* s     = v + ND;
  float* agg   = s + ND;
  float* h     = agg + ND;
  float* e     = h + ND;
  float* alpha = e + ED;
  float* denom = alpha + EH;
  int*   amax  = (int*)(denom + NH);
  float* f1 = q;  // q/k dead after k_edge; reuse for FFN activations
  float* f2 = k;
  float* out = (float*)d_out;

  const int initGrid = (int)((ND + 255) / 256);
  k_init<<<initGrid, 256, 0, stream>>>(amax, denom, agg, N);

  // Node-side fused QKV + skip GEMM (weight-stationary WMMA fp32)
  {
    dim3 grid((unsigned)((rowTiles < 512) ? rowTiles : 512), 4);
    k_qkvs<<<grid, 128, 0, stream>>>(x, Wq, bq, Wk, bk, Wv, bv, Ws, bs, q, k, v,
                                     s, rowTiles);
  }

  // Edge GEMM + fused attention logits
  {
    int gx = (edgeTiles < 1024) ? edgeTiles : 1024;
    k_edge<<<gx, 128, 0, stream>>>(ei, eattr, We, q, k, e, alpha, E, edgeTiles);
  }

  // Scatter softmax
  const int ehGrid = (int)((EH + 255) / 256);
  k_amax<<<ehGrid, 256, 0, stream>>>(alpha, ei, amax, E);
  k_softmax<<<ehGrid, 256, 0, stream>>>(alpha, ei, amax, denom, E);

  // Weighted message scatter
  const int edGrid = (int)((ED + 255) / 256);
  k_msg<<<edGrid, 256, 0, stream>>>(alpha, denom, ei, v, e, agg, E);

  // h = x + LN(agg + skip)
  const int lnGrid = (N + 7) / 8;
  k_ln<true><<<lnGrid, 256, 0, stream>>>(x, agg, s, g1, be1, h, N);

  // FFN
  {
    int gx = (rowTiles < 512) ? rowTiles : 512;
    k_ffn<true><<<gx, 128, 0, stream>>>(h, W1, b1, f1, rowTiles);
    k_ffn<false><<<gx, 128, 0, stream>>>(f1, W2, b2, f2, rowTiles);
  }

  // out = h + LN(f2)
  k_ln<false><<<lnGrid, 256, 0, stream>>>(h, f2, nullptr, g2, be2, out, N);
}